// BasicMFNet_89902255440462
// MI455X (gfx1250) — compile-verified
//
#include <hip/hip_runtime.h>

#define NUM_USER   8192
#define NUM_ITEM   4096
#define NUM_HIDDEN 128
#define NNZ        500000

typedef __attribute__((ext_vector_type(2))) float v2f;
typedef __attribute__((ext_vector_type(8))) float v8f;

// ---------------------------------------------------------------------------
// Kernel 1: zero the mask/pred region and the label region of d_out
// (2 * 8192*4096 floats), and write the trailing ratio scalar.
// ---------------------------------------------------------------------------
__global__ __launch_bounds__(256)
void mf_zero_kernel(float4* __restrict__ p, int n4, float* ratio_ptr, float ratio) {
    int i      = blockIdx.x * blockDim.x + threadIdx.x;
    int stride = gridDim.x * blockDim.x;
    float4 z = make_float4(0.f, 0.f, 0.f, 0.f);
    for (; i < n4; i += stride) p[i] = z;
    if (blockIdx.x == 0 && threadIdx.x == 0) *ratio_ptr = ratio;
}

// ---------------------------------------------------------------------------
// Kernel 2: COO scatter-add. mask is staged IN-PLACE in the mask*pred output
// region (overwritten by the GEMM kernel afterwards); label goes to its slot.
// ---------------------------------------------------------------------------
__global__ __launch_bounds__(256)
void mf_scatter_kernel(const int* __restrict__ idx, const float* __restrict__ ratings,
                       float* mask, float* label) {
    int i = blockIdx.x * blockDim.x + threadIdx.x;
    if (i >= NNZ) return;
    int r = idx[i];
    int c = idx[NNZ + i];
    long off = (long)r * NUM_ITEM + c;
    atomicAdd(&mask[off], 1.0f);
    atomicAdd(&label[off], ratings[i]);
}

// ---------------------------------------------------------------------------
// Kernel 3: per-wave 16x16 output tile, C = U * V^T via V_WMMA_F32_16X16X4_F32
// (K=128 -> 32 WMMA issues), fused with the mask multiply, stored in place.
// Tiles whose 16x16 mask block is entirely zero skip the WMMA loop.
// 256 threads = 8 wave32 per block, one tile per wave.
// ---------------------------------------------------------------------------
__global__ __launch_bounds__(256)
void mf_wmma_gemm_mask_kernel(const float* __restrict__ U, const float* __restrict__ V,
                              float* out) {
    const int wave  = threadIdx.x >> 5;
    const int lane  = threadIdx.x & 31;
    const int half  = lane >> 4;   // 0: lanes 0-15, 1: lanes 16-31
    const int row16 = lane & 15;

    const int tilesN = NUM_ITEM / 16;                    // 256
    long tile = (long)blockIdx.x * 8 + wave;             // 131072 tiles total
    const int tn = (int)(tile % tilesN);
    const int tm = (int)(tile / tilesN);
    const int m0 = tm * 16;
    const int n0 = tn * 16;
    const int col = n0 + row16;                          // N index for C/D layout

    // --- Read the mask tile (currently resident in `out`), C/D VGPR layout:
    // VGPR v, half h -> M = m0 + v + 8*h, N = col.
    float mvals[8];
    bool any = false;
#pragma unroll
    for (int v = 0; v < 8; ++v) {
        int mr = m0 + v + half * 8;
        mvals[v] = out[(long)mr * NUM_ITEM + col];
        any |= (mvals[v] != 0.0f);
    }

    if (!__any((int)any)) {          // uniform across the wave -> no divergence
#pragma unroll
        for (int v = 0; v < 8; ++v) {
            int mr = m0 + v + half * 8;
            out[(long)mr * NUM_ITEM + col] = 0.0f;
        }
        return;
    }

    // --- WMMA dot-product loop. A 16x4 f32 layout: lane row = M, VGPR0/1 hold
    // K = 2*half, 2*half+1 (8-byte contiguous load). B 4x16: lane = N column
    // of V^T = row of V, same K pair. C[m,n] = sum_k U[m,k] * V[n,k].
    const float* arow = U + (long)(m0 + row16) * NUM_HIDDEN;
    const float* brow = V + (long)(n0 + row16) * NUM_HIDDEN;

    v8f c = {};
#pragma unroll
    for (int k0 = 0; k0 < NUM_HIDDEN; k0 += 4) {
        v2f a = *(const v2f*)(arow + k0 + 2 * half);
        v2f b = *(const v2f*)(brow + k0 + 2 * half);
        // (neg_a, A, neg_b, B, c_mod, C, reuse_a, reuse_b)
        c = __builtin_amdgcn_wmma_f32_16x16x4_f32(false, a, false, b,
                                                  (short)0, c, false, false);
    }

    // --- Fused mask multiply, in-place store over the staged mask.
#pragma unroll
    for (int v = 0; v < 8; ++v) {
        int mr = m0 + v + half * 8;
        out[(long)mr * NUM_ITEM + col] = mvals[v] * c[v];
    }
}

// ---------------------------------------------------------------------------
extern "C" void kernel_launch(void* const* d_in, const int* in_sizes, int n_in,
                              void* d_out, int out_size, void* d_ws, size_t ws_size,
                              hipStream_t stream) {
    const int*   idx     = (const int*)d_in[0];    // [2, NNZ] int32
    const float* ratings = (const float*)d_in[1];  // [NNZ]
    const float* U       = (const float*)d_in[2];  // [8192, 128]
    const float* V       = (const float*)d_in[3];  // [4096, 128]

    float* out      = (float*)d_out;
    float* maskpred = out;                                   // [8192, 4096]
    float* label    = out + (long)NUM_USER * NUM_ITEM;       // [8192, 4096]
    float* ratio_p  = out + 2L * NUM_USER * NUM_ITEM;        // scalar

    const float ratio = (float)((double)NUM_USER * NUM_ITEM / (double)NNZ);

    // 1) zero mask+label regions (2 * 33.5M floats = 16.7M float4), set ratio
    const int n4 = (2 * NUM_USER * NUM_ITEM) / 4;
    mf_zero_kernel<<<4096, 256, 0, stream>>>((float4*)out, n4, ratio_p, ratio);

    // 2) scatter-add mask (in-place in pred region) and label
    mf_scatter_kernel<<<(NNZ + 255) / 256, 256, 0, stream>>>(idx, ratings, maskpred, label);

    // 3) WMMA GEMM fused with mask multiply; 131072 tiles / 8 waves per block
    const int tiles  = (NUM_USER / 16) * (NUM_ITEM / 16);
    mf_wmma_gemm_mask_kernel<<<tiles / 8, 256, 0, stream>>>(U, V, maskpred);
}